// NR_all_GraphAttention1_v2_72258529788588
// MI455X (gfx1250) — compile-verified
//
#include <hip/hip_runtime.h>
#include <math.h>

// ---------------------------------------------------------------------------
// Graph attention (NR_all_GraphAttention1) for MI455X / gfx1250 (wave32).
//
// Design notes (compile-only; reasoning from MI455X specs):
//  * All hot data (feats 25.6MB, rel_emb 0.5MB, edge arrays ~12MB) fits in the
//    192MB L2 -> gather/L2-bound workload. HBM traffic ~0.1GB, trivial at
//    23.3 TB/s. So: coalesced 512B/wave row gathers, no atomics, CSR +
//    wave-per-node segmented softmax/reduction via shfl_xor (wave32).
//  * The only real GEMM is feats[50000x128] @ Wc[128x4] (collapsed attention/
//    gate projections). Done with V_WMMA_F32_16X16X4_F32 (exact f32) chained
//    over K, one wave per 16-node tile. B is zero-padded to 128x16 so the
//    inner loop is branch-free: one 64-bit A load, two 32-bit B loads + v_wmma per K-step.
// ---------------------------------------------------------------------------

#define N_NODES 50000
#define N_REL   1000
#define N_EDGES 600000
#define DIM     128
#define OUTC    384   // 3*DIM, output row width per branch

static_assert(N_NODES % 16 == 0, "k_proj assumes full 16-node tiles");

typedef float  v2f  __attribute__((ext_vector_type(2)));
typedef float  v8f  __attribute__((ext_vector_type(8)));
typedef _Float16 v16h __attribute__((ext_vector_type(16)));

__device__ __forceinline__ float wred_sum(float v) {
  for (int m = 16; m >= 1; m >>= 1) v += __shfl_xor(v, m, 32);
  return v;
}
__device__ __forceinline__ float wred_max(float v) {
  for (int m = 16; m >= 1; m >>= 1) v = fmaxf(v, __shfl_xor(v, m, 32));
  return v;
}
__device__ __forceinline__ float sigmoidf(float x) {
  return 1.0f / (1.0f + __expf(-x));
}

// --- per-relation stats: norm, and dot with attn_kernels[0],[1] -------------
__global__ void k_rel_stats(const float* __restrict__ rel_emb,
                            const float* __restrict__ attn_kernels, // [2*128]
                            float* __restrict__ rel_norm,
                            float* __restrict__ reldot) {           // [R*2]
  int wave = (blockIdx.x * blockDim.x + threadIdx.x) >> 5;
  int lane = threadIdx.x & 31;
  if (wave >= N_REL) return;
  const float* row = rel_emb + (size_t)wave * DIM;
  float ss = 0.f, d0 = 0.f, d1 = 0.f;
  #pragma unroll
  for (int j = 0; j < 4; ++j) {
    float v = row[lane + 32 * j];
    ss += v * v;
    d0 += v * attn_kernels[lane + 32 * j];
    d1 += v * attn_kernels[DIM + lane + 32 * j];
  }
  ss = wred_sum(ss); d0 = wred_sum(d0); d1 = wred_sum(d1);
  if (lane == 0) {
    rel_norm[wave]       = sqrtf(ss);
    reldot[2 * wave + 0] = d0;
    reldot[2 * wave + 1] = d1;
  }
}

// --- collapse attn_w / gate_w into zero-padded 128x16 Wcp -------------------
// logit_attn(e) = <f_dst, w0+w2> + <f_src, w1-w2> + b   (w = attn_w[0:384])
// Columns: 0 = attn-dst, 1 = attn-src, 2 = gate-dst, 3 = gate-src, 4..15 = 0.
__global__ void k_wcomb(const float* __restrict__ aw,
                        const float* __restrict__ gw,
                        float* __restrict__ Wcp) {       // [128][16]
  int k = threadIdx.x;
  if (k >= DIM) return;
  #pragma unroll
  for (int n = 4; n < 16; ++n) Wcp[16 * k + n] = 0.f;
  Wcp[16 * k + 0] = aw[k]       + aw[256 + k];
  Wcp[16 * k + 1] = aw[128 + k] - aw[256 + k];
  Wcp[16 * k + 2] = gw[k]       + gw[256 + k];
  Wcp[16 * k + 3] = gw[128 + k] - gw[256 + k];
}

// --- per-edge prep: int32 src/rel, normalization scale ----------------------
// tri_rel[e] = r_val[e]*rel_emb[r]; normalized = rel_emb[r] * s_e with
// s_e = r_val / max(r_val*||rel_r||, 1e-12)   (r_index[0] is arange(E))
__global__ void k_edge_prep(const long long* __restrict__ adj,      // [2][E]
                            const long long* __restrict__ r_index,  // [2][E]
                            const float* __restrict__ r_val,
                            const float* __restrict__ rel_norm,
                            int* __restrict__ src32,
                            int* __restrict__ rel32,
                            float* __restrict__ s_e) {
  int e = blockIdx.x * blockDim.x + threadIdx.x;
  if (e >= N_EDGES) return;
  int src = (int)adj[(size_t)N_EDGES + e];
  int rel = (int)r_index[(size_t)N_EDGES + e];
  src32[e] = src;
  rel32[e] = rel;
  float rv  = r_val[e];
  float nrm = rv * rel_norm[rel];
  s_e[e] = rv / fmaxf(nrm, 1e-12f);
}

// --- CSR row pointers from sorted dst (binary search per node) --------------
__global__ void k_row_ptr(const long long* __restrict__ dst, // adj row 0
                          int* __restrict__ row_ptr) {       // [N+1]
  int v = blockIdx.x * blockDim.x + threadIdx.x;
  if (v > N_NODES) return;
  int lo = 0, hi = N_EDGES;
  while (lo < hi) {
    int mid = (lo + hi) >> 1;
    if (dst[mid] < (long long)v) lo = mid + 1; else hi = mid;
  }
  row_ptr[v] = lo;
}

// --- layer-0 features: tanh(features) into col 0:128 of both outputs --------
__global__ void k_tanh_init(const float* __restrict__ features,
                            float* __restrict__ out_c,
                            float* __restrict__ out_s) {
  int i = blockIdx.x * blockDim.x + threadIdx.x;
  if (i >= N_NODES * DIM) return;
  int n = i >> 7, k = i & 127;
  float t = tanhf(features[i]);
  out_c[(size_t)n * OUTC + k] = t;
  out_s[(size_t)n * OUTC + k] = t;
}

// --- structural pass: wave per node, lanes stripe the 128-dim ---------------
__global__ void k_struct(float* __restrict__ out_c,          // read col l*128, write (l+1)*128
                         float* __restrict__ attst,          // [E] struct attention out
                         const int* __restrict__ row_ptr,
                         const int* __restrict__ src32,
                         const int* __restrict__ rel32,
                         const float* __restrict__ s_e,
                         const float* __restrict__ reldot,
                         const float* __restrict__ rel_emb,
                         int layer) {
  int wave = (blockIdx.x * blockDim.x + threadIdx.x) >> 5;
  int lane = threadIdx.x & 31;
  if (wave >= N_NODES) return;
  int e0 = row_ptr[wave], e1 = row_ptr[wave + 1];
  const float* fin = out_c + (size_t)layer * DIM;  // row stride OUTC

  // segment softmax of score_e = s_e * <rel_emb[r], kernel_l>
  float m = -INFINITY;
  for (int b = e0; b < e1; b += 32) {
    int e = b + lane;
    if (e < e1) m = fmaxf(m, s_e[e] * reldot[2 * rel32[e] + layer]);
  }
  m = wred_max(m);
  float ssum = 0.f;
  for (int b = e0; b < e1; b += 32) {
    int e = b + lane;
    if (e < e1) ssum += __expf(s_e[e] * reldot[2 * rel32[e] + layer] - m);
  }
  ssum = wred_sum(ssum);
  float inv = (ssum > 0.f) ? 1.f / ssum : 0.f;

  // accumulate reflected neighbors; lane owns feature elems 4*lane..4*lane+3
  float4 acc = make_float4(0.f, 0.f, 0.f, 0.f);
  for (int e = e0; e < e1; ++e) {
    int   srcn = src32[e];
    int   rel  = rel32[e];
    float s    = s_e[e];
    float w    = __expf(s * reldot[2 * rel + layer] - m) * inv;
    float4 h = *(const float4*)(fin + (size_t)srcn * OUTC + lane * 4);
    float4 u = *(const float4*)(rel_emb + (size_t)rel * DIM + lane * 4);
    u.x *= s; u.y *= s; u.z *= s; u.w *= s;
    float d = wred_sum(h.x * u.x + h.y * u.y + h.z * u.z + h.w * u.w);
    float c2d = 2.f * d;
    acc.x += w * (h.x - c2d * u.x);
    acc.y += w * (h.y - c2d * u.y);
    acc.z += w * (h.z - c2d * u.z);
    acc.w += w * (h.w - c2d * u.w);
    if (lane == 0) attst[e] = w;
  }
  float4 r = make_float4(tanhf(acc.x), tanhf(acc.y), tanhf(acc.z), tanhf(acc.w));
  *(float4*)(out_c + (size_t)wave * OUTC + (size_t)(layer + 1) * DIM + lane * 4) = r;
}

// --- projection GEMM via WMMA: P[n][0..3] = feats_new[n] @ Wc (+bias) -------
// One wave per 16-node tile (50000 = 3125 * 16 exactly -> no tail guards).
// Branch-free inner loop: A fragment = global_load_b64 of the node row,
// B fragment = two 32-bit loads (rows ka, ka+1) of the zero-padded 128x16 weight panel.
__global__ void k_proj(const float* __restrict__ out_c,
                       const float* __restrict__ Wcp,    // [128][16], cols>=4 zero
                       const float* __restrict__ ab,     // attn_b
                       const float* __restrict__ gb,     // gate_b
                       float* __restrict__ P,            // [N][4]
                       int layer) {
  int wave = (blockIdx.x * blockDim.x + threadIdx.x) >> 5;
  int lane = threadIdx.x & 31;
  if (wave * 16 >= N_NODES) return;
  int half = lane >> 4;     // K-group / row-group select
  int lid  = lane & 15;
  const float* fin = out_c + (size_t)(layer + 1) * DIM;  // row stride OUTC

  v8f c = {0.f, 0.f, 0.f, 0.f, 0.f, 0.f, 0.f, 0.f};
  const float* arow = fin + (size_t)(wave * 16 + lid) * OUTC;

  // A 16x4 per step: lane holds M=lid, VGPR k-pair base = 2*half.
  // B 4x16 per step: lane holds N=lid, same K striping (padded cols read 0).
  for (int kk = 0; kk < DIM; kk += 4) {
    int ka = kk + 2 * half;
    v2f a = *(const v2f*)(arow + ka);
    v2f b;
    b.x = Wcp[(size_t)ka * 16 + lid];
    b.y = Wcp[(size_t)(ka + 1) * 16 + lid];
    c = __builtin_amdgcn_wmma_f32_16x16x4_f32(false, a, false, b,
                                              (short)0, c, false, false);
  }

  // D layout: lane group half -> rows j+8*half, N = lid. Keep N<4.
  if (lid < 4) {
    float bias = (lid == 0) ? ab[0] : ((lid == 2) ? gb[0] : 0.f);
    #pragma unroll
    for (int j = 0; j < 8; ++j) {
      int mm = wave * 16 + j + 8 * half;
      P[(size_t)mm * 4 + lid] = c[j] + bias;
    }
  }
}

// --- semantic pass: wave per node ------------------------------------------
__global__ void k_semantic(const float* __restrict__ out_c,
                           float* __restrict__ out_s,
                           const int* __restrict__ row_ptr,
                           const int* __restrict__ src32,
                           const float* __restrict__ attst,
                           const float* __restrict__ P,   // [N][4]
                           int layer) {
  int wave = (blockIdx.x * blockDim.x + threadIdx.x) >> 5;
  int lane = threadIdx.x & 31;
  if (wave >= N_NODES) return;
  int e0 = row_ptr[wave], e1 = row_ptr[wave + 1];
  const float* fin = out_c + (size_t)(layer + 1) * DIM;   // row stride OUTC
  float pd0 = P[(size_t)wave * 4 + 0];   // dst attn logit part (incl. bias)
  float pd2 = P[(size_t)wave * 4 + 2];   // dst gate logit part (incl. bias)

  // segment softmax of final_att = g*att_score + (1-g)*att_struct
  float m = -INFINITY;
  for (int b = e0; b < e1; b += 32) {
    int e = b + lane;
    if (e < e1) {
      int s = src32[e];
      float4 ps = *(const float4*)(P + (size_t)s * 4);
      float as = fmaxf(sigmoidf(pd0 + ps.y), 1e-4f);
      float g  = sigmoidf(pd2 + ps.w);
      m = fmaxf(m, g * as + (1.f - g) * attst[e]);
    }
  }
  m = wred_max(m);
  float ssum = 0.f;
  for (int b = e0; b < e1; b += 32) {
    int e = b + lane;
    if (e < e1) {
      int s = src32[e];
      float4 ps = *(const float4*)(P + (size_t)s * 4);
      float as = fmaxf(sigmoidf(pd0 + ps.y), 1e-4f);
      float g  = sigmoidf(pd2 + ps.w);
      ssum += __expf(g * as + (1.f - g) * attst[e] - m);
    }
  }
  ssum = wred_sum(ssum);
  float inv = (ssum > 0.f) ? 1.f / ssum : 0.f;

  float4 fd = *(const float4*)(fin + (size_t)wave * OUTC + lane * 4);
  float4 acc = make_float4(0.f, 0.f, 0.f, 0.f);
  for (int e = e0; e < e1; ++e) {
    int s = src32[e];
    float4 ps = *(const float4*)(P + (size_t)s * 4);
    float as = fmaxf(sigmoidf(pd0 + ps.y), 1e-4f);
    float g  = sigmoidf(pd2 + ps.w);
    float a  = __expf(g * as + (1.f - g) * attst[e] - m) * inv;
    float4 h = *(const float4*)(fin + (size_t)s * OUTC + lane * 4);
    acc.x += a * (fd.x - h.x);
    acc.y += a * (fd.y - h.y);
    acc.z += a * (fd.z - h.z);
    acc.w += a * (fd.w - h.w);
  }
  float4 r = make_float4(tanhf(acc.x), tanhf(acc.y), tanhf(acc.z), tanhf(acc.w));
  *(float4*)(out_s + (size_t)wave * OUTC + (size_t)(layer + 1) * DIM + lane * 4) = r;
}

// ---------------------------------------------------------------------------
extern "C" void kernel_launch(void* const* d_in, const int* in_sizes, int n_in,
                              void* d_out, int out_size, void* d_ws, size_t ws_size,
                              hipStream_t stream) {
  (void)in_sizes; (void)n_in; (void)out_size; (void)ws_size;

  const float*     features     = (const float*)d_in[0];
  const float*     rel_emb      = (const float*)d_in[1];
  const long long* adj          = (const long long*)d_in[2];  // int64 [2][E]
  const long long* r_index      = (const long long*)d_in[3];  // int64 [2][E]
  const float*     r_val        = (const float*)d_in[4];
  const float*     attn_w       = (const float*)d_in[5];      // [384]
  const float*     attn_b       = (const float*)d_in[6];      // [1]
  const float*     gate_w       = (const float*)d_in[7];      // [384]
  const float*     gate_b       = (const float*)d_in[8];      // [1]
  const float*     attn_kernels = (const float*)d_in[9];      // [2*128]

  // workspace carve-up (~10.7 MB total, 256B-aligned chunks)
  char*  ws  = (char*)d_ws;
  size_t off = 0;
  auto take = [&](size_t bytes) -> char* {
    char* p = ws + off;
    off = (off + bytes + 255) & ~(size_t)255;
    return p;
  };
  float* rel_norm = (float*)take((size_t)N_REL * 4);
  float* reldot   = (float*)take((size_t)N_REL * 2 * 4);
  float* Wcp      = (float*)take((size_t)DIM * 16 * 4);
  float* s_e      = (float*)take((size_t)N_EDGES * 4);
  int*   src32    = (int*)  take((size_t)N_EDGES * 4);
  int*   rel32    = (int*)  take((size_t)N_EDGES * 4);
  int*   row_ptr  = (int*)  take((size_t)(N_NODES + 1) * 4);
  float* attst    = (float*)take((size_t)N_EDGES * 4);
  float* P        = (float*)take((size_t)N_NODES * 4 * 4);

  float* out_c = (float*)d_out;                        // [N][384]
  float* out_s = out_c + (size_t)N_NODES * OUTC;       // [N][384]

  k_rel_stats<<<(N_REL * 32 + 255) / 256, 256, 0, stream>>>(
      rel_emb, attn_kernels, rel_norm, reldot);
  k_wcomb<<<1, 128, 0, stream>>>(attn_w, gate_w, Wcp);
  k_edge_prep<<<(N_EDGES + 255) / 256, 256, 0, stream>>>(
      adj, r_index, r_val, rel_norm, src32, rel32, s_e);
  k_row_ptr<<<(N_NODES + 1 + 255) / 256, 256, 0, stream>>>(adj, row_ptr);
  k_tanh_init<<<(N_NODES * DIM + 255) / 256, 256, 0, stream>>>(
      features, out_c, out_s);

  for (int l = 0; l < 2; ++l) {
    k_struct<<<(N_NODES * 32 + 255) / 256, 256, 0, stream>>>(
        out_c, attst, row_ptr, src32, rel32, s_e, reldot, rel_emb, l);
    k_proj<<<((N_NODES / 16) * 32 + 255) / 256, 256, 0, stream>>>(
        out_c, Wcp, attn_b, gate_b, P, l);
    k_semantic<<<(N_NODES * 32 + 255) / 256, 256, 0, stream>>>(
        out_c, out_s, row_ptr, src32, attst, P, l);
  }
}